// CustomGraphSAGEModel_69638599737518
// MI455X (gfx1250) — compile-verified
//
#include <hip/hip_runtime.h>
#include <hip/hip_bf16.h>

// GraphSAGE pipeline for MI455X (gfx1250, wave32).
// f32 WMMA (V_WMMA_F32_16X16X4_F32): one 16x16 output tile per wave.
// K and NC are compile-time so all fragment loads use immediate offsets.

static constexpr int kN   = 50000;   // nodes (divisible by 16 -> 3125 M-tiles)
static constexpr int kE   = 640000;  // edges
static constexpr int kOut = 64;

typedef __attribute__((ext_vector_type(2))) float v2f;
typedef __attribute__((ext_vector_type(8))) float v8f;

// ---------------- degree / inv_deg ----------------

__global__ void degree_kernel(const int* __restrict__ dst, float* __restrict__ deg, int E) {
  int e = blockIdx.x * blockDim.x + threadIdx.x;
  if (e < E) unsafeAtomicAdd(&deg[dst[e]], 1.0f);
}

__global__ void invdeg_kernel(float* __restrict__ deg, int n) {
  int i = blockIdx.x * blockDim.x + threadIdx.x;
  if (i < n) {
    float d = deg[i];
    deg[i] = (d > 0.0f) ? (1.0f / d) : 0.0f;
  }
}

// ---------------- mean-aggregation scatter (one wave per edge) ----------------

__global__ void aggregate_kernel(const float* __restrict__ x,
                                 const int* __restrict__ src,
                                 const int* __restrict__ dst,
                                 float* __restrict__ agg, int E, int C) {
  int w = blockIdx.x * (blockDim.x >> 5) + (threadIdx.x >> 5);
  if (w >= E) return;
  int lane = threadIdx.x & 31;
  const float* xr = x + (long long)src[w] * C;
  float* ar = agg + (long long)dst[w] * C;
  for (int c = (lane << 2); c < C; c += 128) {
    float4 v = *(const float4*)(xr + c);
    unsafeAtomicAdd(ar + c + 0, v.x);
    unsafeAtomicAdd(ar + c + 1, v.y);
    unsafeAtomicAdd(ar + c + 2, v.z);
    unsafeAtomicAdd(ar + c + 3, v.w);
  }
}

// ---------------- fused WMMA GEMM ----------------
// acc += A[row0:row0+16, :] @ W[:, col0:col0+16], K/NC compile-time.
// A fragment (16x4 f32): lanes 0-15 K={k,k+1}, lanes 16-31 K={k+2,k+3}, M=lane&15.
// B fragment (4x16):     lanes 0-15 rows k,k+1; lanes 16-31 rows k+2,k+3; N=lane&15.
// C/D (16x16): VGPR v -> M = v + 8*(lane>=16), N = lane&15.

template <int K, int NC, bool SCALE>
__device__ inline v8f wmma_pair(v8f acc, const float* __restrict__ A,
                                const float* __restrict__ W,
                                const float* __restrict__ S,
                                int row0, int col, int lane) {
  const int m  = lane & 15;
  const int hi = lane >> 4;
  const float* ap = A + (long long)(row0 + m) * K + (hi << 1);
  const float* bp = W + (hi << 1) * NC + col;
  float s = 1.0f;
  if (SCALE) s = S[row0 + m];
#pragma unroll
  for (int k = 0; k < K; k += 4) {   // compile-time k -> immediate load offsets
    v2f a;
    a.x = ap[k];
    a.y = ap[k + 1];
    if (SCALE) { a.x *= s; a.y *= s; }
    v2f b;
    b.x = bp[k * NC];
    b.y = bp[(k + 1) * NC];
    acc = __builtin_amdgcn_wmma_f32_16x16x4_f32(false, a, false, b, (short)0, acc,
                                                false, false);
  }
  return acc;
}

template <int K, int NC>
__global__ __launch_bounds__(256) void sage_gemm_kernel(
    const float* __restrict__ A0, const float* __restrict__ W0, const float* __restrict__ S0,
    const float* __restrict__ A1, const float* __restrict__ W1,
    const float* __restrict__ A2, const float* __restrict__ W2,
    const float* __restrict__ bias0, const float* __restrict__ bias1,
    float* __restrict__ out, int M, int ldOut, int relu) {
  constexpr int ntiles = NC / 16;
  const int total = (M >> 4) * ntiles;
  const int wid = blockIdx.x * (blockDim.x >> 5) + (threadIdx.x >> 5);
  if (wid >= total) return;  // wave-uniform: EXEC all-ones around WMMA
  const int row0 = (wid / ntiles) << 4;
  const int col0 = (wid % ntiles) << 4;
  const int lane = threadIdx.x & 31;
  const int col  = col0 + (lane & 15);

  v8f acc = {};
  acc = wmma_pair<K, NC, true>(acc, A0, W0, S0, row0, col, lane);
  if (A1) acc = wmma_pair<K, NC, false>(acc, A1, W1, nullptr, row0, col, lane);
  if (A2) acc = wmma_pair<K, NC, false>(acc, A2, W2, nullptr, row0, col, lane);

  float bv = 0.0f;
  if (bias0) bv += bias0[col];
  if (bias1) bv += bias1[col];
  const int hi = lane >> 4;
#pragma unroll
  for (int v = 0; v < 8; v++) {
    float val = acc[v] + bv;
    if (relu) val = fmaxf(val, 0.0f);
    out[(long long)(row0 + (hi << 3) + v) * ldOut + col] = val;
  }
}

// ---------------- row log-softmax over 64 cols (one wave per row) ----------------

__global__ void logsoftmax64_kernel(const float* __restrict__ in, float* __restrict__ out,
                                    int M) {
  int row = blockIdx.x * (blockDim.x >> 5) + (threadIdx.x >> 5);
  if (row >= M) return;
  int lane = threadIdx.x & 31;
  const float* r = in + (long long)row * 64;
  float a = r[lane];
  float b = r[lane + 32];
  float mx = fmaxf(a, b);
#pragma unroll
  for (int off = 16; off > 0; off >>= 1) mx = fmaxf(mx, __shfl_xor(mx, off, 32));
  float s = __expf(a - mx) + __expf(b - mx);
#pragma unroll
  for (int off = 16; off > 0; off >>= 1) s += __shfl_xor(s, off, 32);
  float lse = __logf(s) + mx;
  out[(long long)row * 64 + lane]      = a - lse;
  out[(long long)row * 64 + lane + 32] = b - lse;
}

// ---------------- host orchestration ----------------

extern "C" void kernel_launch(void* const* d_in, const int* in_sizes, int n_in,
                              void* d_out, int out_size, void* d_ws, size_t ws_size,
                              hipStream_t stream) {
  (void)in_sizes; (void)n_in; (void)out_size; (void)ws_size;

  const float* x     = (const float*)d_in[0];
  const int*   edges = (const int*)d_in[1];
  const int*   src   = edges;
  const int*   dst   = edges + kE;
  const float* f_Wl  = (const float*)d_in[18];
  const float* f_Wr  = (const float*)d_in[19];
  const float* f_b   = (const float*)d_in[20];

  // Workspace layout (256B aligned, all sizes multiples of 256):
  // invdeg | comb[N,256] | hA[N,128] | agg[N,128] | hB[N,128]
  // aggC[N,256] aliases contiguous (agg|hB); outpre[N,64] aliases hA.
  char* ws = (char*)d_ws;
  size_t off = 0;
  auto carve = [&](size_t bytes) {
    void* p = ws + off;
    off += (bytes + 255) & ~(size_t)255;
    return p;
  };
  float* invdeg = (float*)carve((size_t)kN * 4);
  float* comb   = (float*)carve((size_t)kN * 256 * 4);
  float* hA     = (float*)carve((size_t)kN * 128 * 4);
  float* agg    = (float*)carve((size_t)kN * 128 * 4);
  float* hB     = (float*)carve((size_t)kN * 128 * 4);
  float* aggC   = agg;  // 51.2 MB contiguous span agg..hB
  float* outpre = hA;   // hA is dead by the time the final layer runs

  const int aggBlocks = (kE + 7) / 8;  // 8 waves/block, 1 edge/wave

  auto run_gemm128 = [&](const float* A0, const float* W0, const float* S0,
                         const float* A1, const float* W1,
                         const float* A2, const float* W2,
                         const float* b0, const float* b1,
                         float* out, int ldOut, int relu) {
    int total  = (kN / 16) * (128 / 16);
    int blocks = (total + 7) / 8;
    sage_gemm_kernel<128, 128><<<blocks, 256, 0, stream>>>(A0, W0, S0, A1, W1, A2, W2,
                                                           b0, b1, out, kN, ldOut, relu);
  };

  // degree -> inv_deg
  hipMemsetAsync(invdeg, 0, (size_t)kN * 4, stream);
  degree_kernel<<<(kE + 255) / 256, 256, 0, stream>>>(dst, invdeg, kE);
  invdeg_kernel<<<(kN + 255) / 256, 256, 0, stream>>>(invdeg, kN);

  for (int c = 0; c < 2; c++) {
    int base = 2 + c * 8;
    const float* W0l  = (const float*)d_in[base + 0];
    const float* W0r  = (const float*)d_in[base + 1];
    const float* b0   = (const float*)d_in[base + 2];
    const float* Wl   = (const float*)d_in[base + 3];  // [3,128,128]
    const float* Wr   = (const float*)d_in[base + 4];
    const float* b    = (const float*)d_in[base + 5];  // [3,128]
    const float* resW = (const float*)d_in[base + 6];
    const float* resb = (const float*)d_in[base + 7];

    // layer 0: h = relu((agg * inv_deg) @ W0l + x @ W0r + b0)
    hipMemsetAsync(agg, 0, (size_t)kN * 128 * 4, stream);
    aggregate_kernel<<<aggBlocks, 256, 0, stream>>>(x, src, dst, agg, kE, 128);
    run_gemm128(agg, W0l, invdeg, x, W0r, nullptr, nullptr, b0, nullptr, hA, 128, 1);

    // layers 1..3: h = (agg*inv_deg)@Wl + h@Wr + h@resW + (b + resb), relu except last
    float* cur = hA;
    float* nxt = hB;
    for (int i = 0; i < 3; i++) {
      hipMemsetAsync(agg, 0, (size_t)kN * 128 * 4, stream);
      aggregate_kernel<<<aggBlocks, 256, 0, stream>>>(cur, src, dst, agg, kE, 128);
      float* out = (i == 2) ? (comb + c * 128) : nxt;  // last layer -> concat slot
      int ldo = (i == 2) ? 256 : 128;
      run_gemm128(agg, Wl + (size_t)i * 128 * 128, invdeg,
                  cur, Wr + (size_t)i * 128 * 128,
                  cur, resW + (size_t)i * 128 * 128,
                  b + (size_t)i * 128, resb + (size_t)i * 128,
                  out, ldo, (i < 2) ? 1 : 0);
      float* t = cur; cur = nxt; nxt = t;
    }
  }

  // final SAGE on combined [N,256] -> [N,64], then log_softmax
  hipMemsetAsync(aggC, 0, (size_t)kN * 256 * 4, stream);
  aggregate_kernel<<<aggBlocks, 256, 0, stream>>>(comb, src, dst, aggC, kE, 256);
  {
    int total  = (kN / 16) * (kOut / 16);
    int blocks = (total + 7) / 8;
    sage_gemm_kernel<256, 64><<<blocks, 256, 0, stream>>>(
        aggC, f_Wl, invdeg, comb, f_Wr, nullptr, nullptr, f_b, nullptr,
        outpre, kN, kOut, 0);
  }
  logsoftmax64_kernel<<<(kN + 7) / 8, 256, 0, stream>>>(outpre, (float*)d_out, kN);
}